// ProbAttention_47562467836501
// MI455X (gfx1250) — compile-verified
//
#include <hip/hip_runtime.h>
#include <hip/hip_bf16.h>
#include <math.h>

// ProbSparse attention for MI455X (gfx1250).
// One block per (b,h). Async global->LDS staging, f32 WMMA for Q*K^T,
// wave-per-row double softmax, bandwidth-bound single pass.

typedef float v2f __attribute__((ext_vector_type(2)));
typedef float v8f __attribute__((ext_vector_type(8)));

#define QK_STRIDE 36   // floats/row in LDS for q/k/v: rows 16B aligned (36*4=144), bank-spread
#define S_STRIDE  68   // floats/row for S: avoids 64-bank column conflicts

// ---------------- threefry2x32 (JAX-exact) ----------------
__device__ __forceinline__ unsigned rotl32(unsigned v, int r) {
    return (v << r) | (v >> (32 - r));
}

__device__ void threefry2x32(unsigned k0, unsigned k1, unsigned x0, unsigned x1,
                             unsigned& o0, unsigned& o1) {
    const unsigned ks2 = k0 ^ k1 ^ 0x1BD11BDAu;
    x0 += k0; x1 += k1;
    const int R1[4] = {13, 15, 26, 6};
    const int R2[4] = {17, 29, 16, 24};
#define TF_RND(r) { x0 += x1; x1 = rotl32(x1, (r)); x1 ^= x0; }
    for (int i = 0; i < 4; ++i) TF_RND(R1[i]);  x0 += k1;  x1 += ks2 + 1u;
    for (int i = 0; i < 4; ++i) TF_RND(R2[i]);  x0 += ks2; x1 += k0  + 2u;
    for (int i = 0; i < 4; ++i) TF_RND(R1[i]);  x0 += k0;  x1 += k1  + 3u;
    for (int i = 0; i < 4; ++i) TF_RND(R2[i]);  x0 += k1;  x1 += ks2 + 4u;
    for (int i = 0; i < 4; ++i) TF_RND(R1[i]);  x0 += ks2; x1 += k0  + 5u;
#undef TF_RND
    o0 = x0; o1 = x1;
}

// Reproduce idx_sample = jax.random.randint(key(42), (64,25), 0, 64).
// key(42) = (0,42); randint splits the key; span=64 is a power of two so
// idx = lower_bits & 63 where lower_bits = random_bits(k2, 32, (64,25)).
__global__ void probattn_idx_kernel(int* __restrict__ idx) {
    int i = blockIdx.x * blockDim.x + threadIdx.x;
    if (i >= 800) return;
    // split(key): threefry((0,42), counts=[0,1,2,3]) with x0=[0,1], x1=[2,3];
    // flat out = [o0(0,2), o0(1,3), o1(0,2), o1(1,3)]; k2 = (out[2], out[3]).
    unsigned a0, a1, b0, b1;
    threefry2x32(0u, 42u, 0u, 2u, a0, a1);
    threefry2x32(0u, 42u, 1u, 3u, b0, b1);
    const unsigned k2_0 = a1, k2_1 = b1;
    // random_bits(k2, 32, 1600): x0 = i, x1 = 800+i; flat = concat(o0_vec, o1_vec)
    unsigned o0, o1;
    threefry2x32(k2_0, k2_1, (unsigned)i, (unsigned)(800 + i), o0, o1);
    idx[i]       = (int)(o0 & 63u);
    idx[800 + i] = (int)(o1 & 63u);
}

// ---------------- main kernel ----------------
__launch_bounds__(128)
__global__ void probattn_main(const float* __restrict__ q, const float* __restrict__ k,
                              const float* __restrict__ v, const float* __restrict__ rpb,
                              const float* __restrict__ swm, const int* __restrict__ idxs,
                              float* __restrict__ out) {
    __shared__ __align__(16) float qs[64 * QK_STRIDE];
    __shared__ __align__(16) float ks_[64 * QK_STRIDE];
    __shared__ __align__(16) float vs[64 * QK_STRIDE];
    __shared__ float S[64 * S_STRIDE];
    __shared__ float Mv[64];
    __shared__ int   sel[64];
    __shared__ int   rowsel[64];
    __shared__ int   nsel_s;
    __shared__ float vmean[32];
    __shared__ float pbuf[4][64];

    const int tid  = threadIdx.x;
    const int wave = tid >> 5;
    const int lane = tid & 31;
    const int bid  = blockIdx.x;
    const int b    = bid >> 3;     // batch
    const int h    = bid & 7;      // head
    const int w    = b & 1023;     // window = b % nW

    // (B,L,H,D) layout: element (b,l,h,d) at ((b*64+l)*8+h)*32+d.
    const size_t baseElem = (size_t)b * 16384 + (size_t)h * 32;
    const float* qb = q + baseElem;
    const float* kb = k + baseElem;
    const float* vb = v + baseElem;

    // ---- async global -> LDS staging of the three 64x32 tiles (8KB each) ----
    const unsigned qsB = (unsigned)(size_t)&qs[0];
    const unsigned ksB = (unsigned)(size_t)&ks_[0];
    const unsigned vsB = (unsigned)(size_t)&vs[0];
#pragma unroll
    for (int i = 0; i < 4; ++i) {
        const int c   = tid + i * 128;       // 512 chunks of 16B per matrix
        const int row = c >> 3;
        const int ch  = c & 7;
        const unsigned goff = (unsigned)(row * 1024 + ch * 16);  // row stride H*D*4 = 1024B
        const unsigned loff = (unsigned)(row * (QK_STRIDE * 4) + ch * 16);
        asm volatile("global_load_async_to_lds_b128 %0, %1, %2"
                     :: "v"(qsB + loff), "v"(goff), "s"(qb) : "memory");
        asm volatile("global_load_async_to_lds_b128 %0, %1, %2"
                     :: "v"(ksB + loff), "v"(goff), "s"(kb) : "memory");
        asm volatile("global_load_async_to_lds_b128 %0, %1, %2"
                     :: "v"(vsB + loff), "v"(goff), "s"(vb) : "memory");
    }
    asm volatile("s_wait_asynccnt 0" ::: "memory");
    __syncthreads();

    // ---- S = Q*K^T via V_WMMA_F32_16X16X4_F32; wave w owns rows [16w,16w+16) ----
    const int row0  = wave * 16;
    const int lsub  = lane & 15;
    const int khalf = (lane < 16) ? 0 : 2;   // A/B 16x4 f32 layout: upper half-wave holds K=2,3
#pragma unroll
    for (int n = 0; n < 4; ++n) {
        const int col0 = n * 16;
        v8f c = {0.f, 0.f, 0.f, 0.f, 0.f, 0.f, 0.f, 0.f};
#pragma unroll
        for (int kb4 = 0; kb4 < 32; kb4 += 4) {
            v2f a, bmat;
            a.x    = qs[(row0 + lsub) * QK_STRIDE + kb4 + khalf];
            a.y    = qs[(row0 + lsub) * QK_STRIDE + kb4 + khalf + 1];
            bmat.x = ks_[(col0 + lsub) * QK_STRIDE + kb4 + khalf];   // B[k][n] = K[n][k]
            bmat.y = ks_[(col0 + lsub) * QK_STRIDE + kb4 + khalf + 1];
            c = __builtin_amdgcn_wmma_f32_16x16x4_f32(false, a, false, bmat,
                                                      (short)0, c, false, false);
        }
        // C/D layout: VGPR r -> M = r (lanes 0-15) / 8+r (lanes 16-31), N = lane%16
        const int rbase = row0 + ((lane < 16) ? 0 : 8);
#pragma unroll
        for (int r = 0; r < 8; ++r)
            S[(rbase + r) * S_STRIDE + col0 + lsub] = c[r];
    }
    __syncthreads();

    // ---- sparsity measure over sampled columns (raw, unscaled S) ----
    if (tid < 64) {
        float mx = -__builtin_inff(), sm = 0.f;
        const int* ip = idxs + tid * 25;
        for (int j = 0; j < 25; ++j) {
            const float x = S[tid * S_STRIDE + ip[j]];
            mx = fmaxf(mx, x);
            sm += x;
        }
        Mv[tid] = mx - sm * (1.0f / 64.0f);
    }
    __syncthreads();

    // ---- top-25 rows (stable ranking; output order is irrelevant) ----
    if (tid < 64) {
        const float mi = Mv[tid];
        int rank = 0;
        for (int j = 0; j < 64; ++j) {
            const float mj = Mv[j];
            rank += (mj > mi) || (mj == mi && j < tid);
        }
        sel[tid] = (rank < 25) ? 1 : 0;
    }
    __syncthreads();
    if (tid == 0) {
        int c = 0;
        for (int r = 0; r < 64; ++r)
            if (sel[r]) rowsel[c++] = r;
        nsel_s = c;
    }
    if (tid >= 64 && tid < 96) {            // mean of V over rows (context default)
        const int d = tid - 64;
        float s = 0.f;
        for (int j = 0; j < 64; ++j) s += vs[j * QK_STRIDE + d];
        vmean[d] = s * (1.0f / 64.0f);
    }
    __syncthreads();

    // ---- per selected row: softmax -> +rpb +mask -> softmax -> row*V ----
    const float scale = 0.17677669529663687f;   // 1/sqrt(32)
    const int nsel = nsel_s;
    for (int s = wave; s < nsel; s += 4) {
        const int row = rowsel[s];
        float x0 = S[row * S_STRIDE + lane] * scale;
        float x1 = S[row * S_STRIDE + lane + 32] * scale;
        float m = fmaxf(x0, x1);
        for (int o = 16; o > 0; o >>= 1) m = fmaxf(m, __shfl_xor(m, o, 32));
        float e0 = __expf(x0 - m), e1 = __expf(x1 - m);
        float ssum = e0 + e1;
        for (int o = 16; o > 0; o >>= 1) ssum += __shfl_xor(ssum, o, 32);
        const float inv = 1.0f / ssum;
        const size_t rpbo = (size_t)(h * 64 + row) * 64;
        const size_t mo   = ((size_t)w * 64 + row) * 64;
        float g0 = e0 * inv + rpb[rpbo + lane]      + swm[mo + lane];
        float g1 = e1 * inv + rpb[rpbo + lane + 32] + swm[mo + lane + 32];
        float m2 = fmaxf(g0, g1);
        for (int o = 16; o > 0; o >>= 1) m2 = fmaxf(m2, __shfl_xor(m2, o, 32));
        float f0 = __expf(g0 - m2), f1 = __expf(g1 - m2);
        float s2 = f0 + f1;
        for (int o = 16; o > 0; o >>= 1) s2 += __shfl_xor(s2, o, 32);
        const float inv2 = 1.0f / s2;
        pbuf[wave][lane]      = f0 * inv2;
        pbuf[wave][lane + 32] = f1 * inv2;
        // lane == output dim d (D=32): acc = sum_j P[j] * V[j][d]
        float acc = 0.f;
#pragma unroll 8
        for (int j = 0; j < 64; ++j)
            acc += pbuf[wave][j] * vs[j * QK_STRIDE + lane];
        out[((size_t)(b * 64 + row) * 8 + h) * 32 + lane] = acc;
    }
    __syncthreads();

    // ---- unselected rows get mean(V) ----
    for (int e = tid; e < 2048; e += 128) {
        const int row = e >> 5, d = e & 31;
        if (!sel[row])
            out[((size_t)(b * 64 + row) * 8 + h) * 32 + d] = vmean[d];
    }
}

extern "C" void kernel_launch(void* const* d_in, const int* in_sizes, int n_in,
                              void* d_out, int out_size, void* d_ws, size_t ws_size,
                              hipStream_t stream) {
    const float* q   = (const float*)d_in[0];
    const float* k   = (const float*)d_in[1];
    const float* v   = (const float*)d_in[2];
    const float* rpb = (const float*)d_in[3];
    const float* swm = (const float*)d_in[4];
    // d_in[5] (attn_mask) unused by the reference math.
    int*   idx = (int*)d_ws;           // 1600 ints
    float* out = (float*)d_out;

    probattn_idx_kernel<<<4, 256, 0, stream>>>(idx);
    probattn_main<<<2048 * 8, 128, 0, stream>>>(q, k, v, rpb, swm, idx, out);
}